// TripletLoss_53747220742325
// MI455X (gfx1250) — compile-verified
//
#include <hip/hip_runtime.h>
#include <stdint.h>

typedef float v2f __attribute__((ext_vector_type(2)));
typedef float v8f __attribute__((ext_vector_type(8)));

#define DIM   512
#define NTRIP 65536

// One wave (32 lanes) handles 16 triplets. Lane layout:
//   m    = lane & 15  -> triplet within group (matrix row M)
//   half = lane >> 4  -> which K-pair of the 16x4 f32 A-layout this lane feeds
// For V_WMMA_F32_16X16X4_F32 the B-operand (V^T) register image equals V
// loaded in A-layout, so both operands are built from the lane's own data.
__global__ __launch_bounds__(256) void triplet_wmma_kernel(
    const float* __restrict__ x,
    const long long* __restrict__ trip,
    float* __restrict__ partials)
{
    const int lane = threadIdx.x & 31;
    const int wave = (int)((blockIdx.x * blockDim.x + threadIdx.x) >> 5);
    const int m    = lane & 15;
    const int half = lane >> 4;

    const long long tb = (long long)(wave * 16 + m) * 3;
    const int i = (int)trip[tb + 0];
    const int j = (int)trip[tb + 1];
    const int k = (int)trip[tb + 2];

    const float* __restrict__ xi = x + (size_t)i * DIM + half * 4;
    const float* __restrict__ xj = x + (size_t)j * DIM + half * 4;
    const float* __restrict__ xk = x + (size_t)k * DIM + half * 4;

    v8f acc = {};  // 16x16 f32 accumulator; only the diagonal is meaningful

#pragma unroll 4
    for (int c = 0; c < DIM; c += 8) {
        float4 a = *(const float4*)(xi + c);   // xi chunk (b128 gather)
        float4 b = *(const float4*)(xj + c);   // xj chunk
        float4 g = *(const float4*)(xk + c);   // xk chunk

        // u = (xi-xj) - (xi-xk) = xk - xj ; v = (xi-xj) + (xi-xk)
        float u0 = g.x - b.x, u1 = g.y - b.y, u2 = g.z - b.z, u3 = g.w - b.w;
        float v0 = 2.0f * a.x - b.x - g.x;
        float v1 = 2.0f * a.y - b.y - g.y;
        float v2 = 2.0f * a.z - b.z - g.z;
        float v3 = 2.0f * a.w - b.w - g.w;

        v2f A1 = {u0, u1}, B1 = {v0, v1};
        v2f A2 = {u2, u3}, B2 = {v2, v3};

        // D = A x B + C, exact fp32. diag(U V^T)[m] accumulates sum (a^2-b^2).
        acc = __builtin_amdgcn_wmma_f32_16x16x4_f32(
            false, A1, false, B1, (short)0, acc, false, false);
        acc = __builtin_amdgcn_wmma_f32_16x16x4_f32(
            false, A2, false, B2, (short)0, acc, false, false);
    }

    // Diagonal extraction: diag[r] at (VGPR r, lane r); diag[r+8] at (VGPR r, lane r+24).
    float d = 0.0f;
#pragma unroll
    for (int r = 0; r < 8; ++r) {
        if ((lane == r) || (lane == r + 24)) d = acc[r];
    }

    // Numerically stable softplus on the 16 active lanes.
    float sp = 0.0f;
    if ((lane < 8) || (lane >= 24)) {
        sp = fmaxf(d, 0.0f) + log1pf(expf(-fabsf(d)));
    }

    // wave32 reduction
#pragma unroll
    for (int off = 16; off > 0; off >>= 1)
        sp += __shfl_xor(sp, off, 32);

    if (lane == 0) partials[wave] = sp;
}

__global__ __launch_bounds__(256) void triplet_reduce_kernel(
    const float* __restrict__ partials, float* __restrict__ out, int n)
{
    __shared__ float smem[8];
    float s = 0.0f;
    for (int idx = threadIdx.x; idx < n; idx += 256) s += partials[idx];

#pragma unroll
    for (int off = 16; off > 0; off >>= 1)
        s += __shfl_xor(s, off, 32);

    const int lane = threadIdx.x & 31;
    const int w    = threadIdx.x >> 5;
    if (lane == 0) smem[w] = s;
    __syncthreads();
    if (w == 0) {
        s = (lane < 8) ? smem[lane] : 0.0f;
#pragma unroll
        for (int off = 4; off > 0; off >>= 1)
            s += __shfl_xor(s, off, 32);
        if (lane == 0) out[0] = s * (1.0f / (float)NTRIP);
    }
}

extern "C" void kernel_launch(void* const* d_in, const int* in_sizes, int n_in,
                              void* d_out, int out_size, void* d_ws, size_t ws_size,
                              hipStream_t stream)
{
    const float*     x    = (const float*)d_in[0];       // (8192, 512) fp32
    const long long* trip = (const long long*)d_in[1];   // (65536, 3) int64
    float* out      = (float*)d_out;                     // 1 float
    float* partials = (float*)d_ws;                      // 4096 floats (16 KB)

    const int nWaves  = NTRIP / 16;          // 4096 waves, 16 triplets each
    const int threads = 256;                 // 8 waves per block
    const int blocks  = nWaves * 32 / threads; // 512 blocks

    triplet_wmma_kernel<<<blocks, threads, 0, stream>>>(x, trip, partials);
    triplet_reduce_kernel<<<1, threads, 0, stream>>>(partials, out, nWaves);
}